// HardOIMLoss_13116830122678
// MI455X (gfx1250) — compile-verified
//
#include <hip/hip_runtime.h>
#include <hip/hip_bf16.h>

#define N_ROWS   4096
#define DIM      256
#define P_LUT    5532
#define Q_CQ     5000
#define C_COLS   10532           // P_LUT + Q_CQ
#define HARD_K   701u            // (HARD_NUM+1)-th largest
#define OIM_SCALE 30.0f
#define LDS_PAD  4               // floats of pad per 256-float row (TDM pad feature)

typedef __attribute__((ext_vector_type(2))) float v2f;
typedef __attribute__((ext_vector_type(8))) float v8f;
typedef __attribute__((ext_vector_type(4))) unsigned int u32x4;
typedef __attribute__((ext_vector_type(8))) unsigned int u32x8;

// ---- order-preserving float <-> uint key (for descending radix select) ----
__device__ __forceinline__ unsigned fkey(float x) {
    unsigned b = __float_as_uint(x);
    return (b & 0x80000000u) ? ~b : (b | 0x80000000u);
}
__device__ __forceinline__ float funkey(unsigned u) {
    unsigned b = (u & 0x80000000u) ? (u ^ 0x80000000u) : ~u;
    return __uint_as_float(b);
}

__global__ __launch_bounds__(256) void oim_zero(float* out) {
    if (threadIdx.x == 0) out[0] = 0.0f;
}

// ============================================================================
// GEMM: proj[lrow, col] = <inputs[row0+lrow,:], table[col,:]> * rel[col]
// A tile (16 x 256 f32) staged into LDS by the Tensor Data Mover with HW
// padding (260-float row stride -> conflict-free ds_load_b64 A fragments).
// fp32 WMMA 16x16x4, K looped over 256. Block: 8 waves, tile 16x128.
// ============================================================================
__global__ __launch_bounds__(256)
void oim_gemm_wmma(const float* __restrict__ inputs,
                   const float* __restrict__ lut,
                   const float* __restrict__ cq,
                   const float* __restrict__ rel,
                   float* __restrict__ proj,
                   int row0) {
    __shared__ float At[16][DIM + LDS_PAD];

    const int tid  = threadIdx.x;
    const int wave = tid >> 5;             // wave32
    const int lane = tid & 31;
    const int half = lane >> 4;            // 0: K pair {0,1}; 1: K pair {2,3}
    const int l16  = lane & 15;

    const int lrow0   = blockIdx.y * 16;   // chunk-local row base
    const int grow0   = row0 + lrow0;      // global row base
    const int colBase = blockIdx.x * 128 + wave * 16;

    // ---- TDM: tensor_load_to_lds of the 16x256 f32 A tile (wave 0 issues) ----
    if (wave == 0) {
        const unsigned long long ga =
            (unsigned long long)(const void*)(inputs + (size_t)grow0 * DIM);
        const unsigned ldsOff = (unsigned)(unsigned long long)&At[0][0]; // LDS byte offset

        u32x4 g0;
        g0[0] = 1u;                                   // count=1, user D#, no gather
        g0[1] = ldsOff;                               // lds_addr (bytes)
        g0[2] = (unsigned)(ga & 0xFFFFFFFFu);         // global_addr[31:0]
        g0[3] = (unsigned)((ga >> 32) & 0x01FFFFFFu)  // global_addr[56:32]
                | (2u << 30);                         // type = 2 ("image")

        u32x8 g1;
        g1[0] = (2u << 16)        // data_size = 2 -> 4 bytes
              | (1u << 20)        // pad_enable
              | (7u << 22)        // pad_interval = 7 -> every 256 DWORDs
              | (3u << 25);       // pad_amount  = 3 -> 4 DWORDs (260-float stride)
        g1[1] = ((unsigned)DIM & 0xFFFFu) << 16;      // tensor_dim0[15:0] = 256
        g1[2] = ((unsigned)N_ROWS & 0xFFFFu) << 16;   // tensor_dim1[15:0] = 4096
        g1[3] = ((unsigned)DIM & 0xFFFFu) << 16;      // tile_dim0 = 256
        g1[4] = 16u;                                  // tile_dim1 = 16 rows
        g1[5] = (unsigned)DIM;                        // tensor_dim0_stride = 256
        g1[6] = 0u;
        g1[7] = 0u;

        asm volatile("tensor_load_to_lds %0, %1"
                     :: "s"(g0), "s"(g1) : "memory");
        __builtin_amdgcn_s_wait_tensorcnt(0);
    }
    __syncthreads();

    // ---- per-lane B row pointer (one table row == one output column) ----
    const int col  = colBase + l16;
    const int colc = (col < C_COLS) ? col : (C_COLS - 1);
    const float* brow = (colc < P_LUT) ? (lut + (size_t)colc * DIM)
                                       : (cq + (size_t)(colc - P_LUT) * DIM);
    __builtin_prefetch(brow, 0, 3);        // global_prefetch_b8 (table is L2-resident)
    __builtin_prefetch(brow + 128, 0, 3);

    // ---- K loop: 64 x V_WMMA_F32_16X16X4_F32 ----
    v8f acc = {};
    const int ko = 2 * half;               // lane-half K offset per ISA A/B layout
    #pragma unroll 8
    for (int kb = 0; kb < DIM; kb += 4) {
        v2f a, b;
        a.x = At[l16][kb + ko];            // A: 16x4, M = l16
        a.y = At[l16][kb + ko + 1];
        b.x = brow[kb + ko];               // B: 4x16, N = l16 (B[k][n] = table[col][k])
        b.y = brow[kb + ko + 1];
        acc = __builtin_amdgcn_wmma_f32_16x16x4_f32(
                  false, a, false, b, (short)0, acc, false, false);
    }

    // ---- write D (lane holds column 'col', rows i + 8*half), scale by rel ----
    if (col < C_COLS) {
        const float r = rel[col];
        #pragma unroll
        for (int i = 0; i < 8; ++i) {
            int m = i + 8 * half;
            proj[(size_t)(lrow0 + m) * C_COLS + col] = acc[i] * r;
        }
    }
}

// ============================================================================
// Fused: exact 701st-largest per row (two 16-bit radix passes, 64K-bin LDS
// histogram) + masked log-sum-exp cross-entropy. Row max folded into pass 1.
// One 256-thread block per row; 3 scans of the row total.
// ============================================================================
#define HBINS 65536

__global__ __launch_bounds__(256)
void oim_select_loss(const float* __restrict__ proj,
                     const int* __restrict__ labels,
                     float* __restrict__ out, int row0) {
    __shared__ unsigned hist[HBINS];       // 256 KB (gfx1250: 320 KB LDS per WG)
    __shared__ unsigned csum[256];
    __shared__ float    red[256];
    __shared__ float    s_pmax, s_bnd;
    __shared__ unsigned s_hi, s_rank;

    const int tid = threadIdx.x;
    const float* p = proj + (size_t)blockIdx.x * C_COLS;
    const int lbl = labels[row0 + blockIdx.x];

    for (int i = tid; i < HBINS; i += 256) hist[i] = 0u;
    __syncthreads();

    // ---- scan 1: hi-16 histogram + row max (row max is always masked) ----
    float mx = -1e30f;
    for (int c = tid; c < C_COLS; c += 256) {
        float v = p[c];
        mx = fmaxf(mx, v);
        atomicAdd(&hist[fkey(v) >> 16], 1u);           // ds_add_u32
    }
    red[tid] = mx;
    __syncthreads();
    for (int s = 128; s > 0; s >>= 1) {
        if (tid < s) red[tid] = fmaxf(red[tid], red[tid + s]);
        __syncthreads();
    }
    if (tid == 0) s_pmax = red[0];

    // ---- two-level suffix scan for rank HARD_K (descending) ----
    {
        unsigned cs = 0;
        for (int b = 0; b < 256; ++b) cs += hist[tid * 256 + b];
        csum[tid] = cs;
    }
    __syncthreads();
    if (tid == 0) {
        unsigned r = HARD_K;
        int T = 255;
        for (; T > 0; --T) { if (csum[T] >= r) break; r -= csum[T]; }
        int B = T * 256 + 255;
        for (; B > T * 256; --B) { unsigned h = hist[B]; if (h >= r) break; r -= h; }
        if (hist[B] < r) r = hist[B];      // clamp (degenerate ties)
        s_hi = (unsigned)B;
        s_rank = r;
    }
    __syncthreads();
    const unsigned hi = s_hi;

    // ---- scan 2: lo-16 histogram among prefix matches ----
    for (int i = tid; i < HBINS; i += 256) hist[i] = 0u;
    __syncthreads();
    for (int c = tid; c < C_COLS; c += 256) {
        unsigned u = fkey(p[c]);
        if ((u >> 16) == hi) atomicAdd(&hist[u & 0xFFFFu], 1u);
    }
    __syncthreads();
    {
        unsigned cs = 0;
        for (int b = 0; b < 256; ++b) cs += hist[tid * 256 + b];
        csum[tid] = cs;
    }
    __syncthreads();
    if (tid == 0) {
        unsigned r = s_rank;
        int T = 255;
        for (; T > 0; --T) { if (csum[T] >= r) break; r -= csum[T]; }
        int B = T * 256 + 255;
        for (; B > T * 256; --B) { unsigned h = hist[B]; if (h >= r) break; r -= h; }
        s_bnd = funkey((hi << 16) | (unsigned)B);
    }
    __syncthreads();
    const float bnd = s_bnd, pmax = s_pmax;

    // ---- scan 3: masked sum of exp(30*(v - pmax)) ----
    float sum = 0.0f;
    for (int c = tid; c < C_COLS; c += 256) {
        float v = p[c];
        if (v >= bnd || c == lbl) sum += __expf(OIM_SCALE * (v - pmax));
    }
    red[tid] = sum;
    __syncthreads();
    for (int s = 128; s > 0; s >>= 1) {
        if (tid < s) red[tid] += red[tid + s];
        __syncthreads();
    }
    if (tid == 0) {
        float logZ = __logf(red[0]) + OIM_SCALE * pmax;
        float tgt  = OIM_SCALE * p[lbl];
        atomicAdd(out, (logZ - tgt) * (1.0f / (float)N_ROWS));
    }
}

// ============================================================================
extern "C" void kernel_launch(void* const* d_in, const int* in_sizes, int n_in,
                              void* d_out, int out_size, void* d_ws, size_t ws_size,
                              hipStream_t stream) {
    const float* inputs = (const float*)d_in[0];
    const int*   labels = (const int*)  d_in[1];
    const float* lut    = (const float*)d_in[2];
    const float* cq     = (const float*)d_in[3];
    const float* rel    = (const float*)d_in[4];
    float* out  = (float*)d_out;
    float* proj = (float*)d_ws;              // chunked [rows, C_COLS] fp32

    long avail   = (long)(ws_size / sizeof(float));
    int  maxRows = (int)(avail / C_COLS);
    maxRows = (maxRows / 16) * 16;
    if (maxRows > N_ROWS) maxRows = N_ROWS;
    if (maxRows < 16)     maxRows = 16;      // minimum tile; ws must cover this

    oim_zero<<<1, 256, 0, stream>>>(out);

    for (int row0 = 0; row0 < N_ROWS; row0 += maxRows) {
        int rows = N_ROWS - row0;
        if (rows > maxRows) rows = maxRows;  // always a multiple of 16 (4096 % 16 == 0)
        dim3 grid((C_COLS + 127) / 128, rows / 16);
        oim_gemm_wmma  <<<grid, 256, 0, stream>>>(inputs, lut, cq, rel, proj, row0);
        oim_select_loss<<<rows, 256, 0, stream>>>(proj, labels, out, row0);
    }
}